// vGIN_16604343566536
// MI455X (gfx1250) — compile-verified
//
#include <hip/hip_runtime.h>

#define Nn 50000
#define Ee 600000
#define Dd 128
#define Gg 128
#define Cc 10
#define BN_EPS 1e-5f

typedef __attribute__((ext_vector_type(2))) float v2f;
typedef __attribute__((ext_vector_type(8))) float v8f;

// ---------------------------------------------------------------------------
__global__ void k_copy4(float* __restrict__ dst, const float* __restrict__ src, int n4) {
    int t = blockIdx.x * blockDim.x + threadIdx.x;
    if (t < n4) ((float4*)dst)[t] = ((const float4*)src)[t];
}

// ---------------------------------------------------------------------------
// Edge scatter-add: agg[dst] += h[src]; one thread per (edge, 4 floats)
__global__ void k_scatter(float* __restrict__ agg, const float* __restrict__ h,
                          const int* __restrict__ src, const int* __restrict__ dst) {
    long long t = (long long)blockIdx.x * blockDim.x + threadIdx.x;
    if (t >= (long long)Ee * 32) return;
    int e = (int)(t >> 5);
    int q = ((int)t & 31) * 4;
    int s = src[e], d = dst[e];
    float4 v = *(const float4*)(h + (size_t)s * Dd + q);
    float* o = agg + (size_t)d * Dd + q;
    atomicAdd(o + 0, v.x);
    atomicAdd(o + 1, v.y);
    atomicAdd(o + 2, v.z);
    atomicAdd(o + 3, v.w);
}

// ---------------------------------------------------------------------------
// WMMA fp32 GEMM: Out[M,Ncols] = act(A)[M,K] @ W[K,Ncols] + bias
// BN template: a = relu(a*in_scale[k] + in_shift[k]) applied on A load.
// Block: 256 threads = 8 waves; each wave: 16 rows x two 16-col tiles.
// Block tile: 128 rows x 32 cols.
// W staged in LDS pair-interleaved: sB[p*96 + 2c + j] = W[2p+j][colBase+c]
//  -> each B fragment is one aligned ds_load_b64 straight into the WMMA
//     register pair; pair-row stride 96 = 32 (mod 64 banks) makes the two
//     half-wave K-phases hit complementary bank halves (conflict-free).
#define PSTRIDE 96
template <bool BN>
__global__ __launch_bounds__(256)
void k_gemm(const float* __restrict__ A, int lda,
            const float* __restrict__ W,
            const float* __restrict__ bias,
            const float* __restrict__ in_scale,
            const float* __restrict__ in_shift,
            float* __restrict__ Out,
            float* __restrict__ colSum, float* __restrict__ colSq,
            int M, int K, int Ncols) {
    __shared__ float sB[128 * PSTRIDE];   // up to K=256 (128 K-pairs), 32 cols
    __shared__ float sSum[32];
    __shared__ float sSq[32];

    int tid  = threadIdx.x;
    int wave = tid >> 5;
    int lane = tid & 31;
    int ln   = lane & 15;      // row (A) / col (B,C)
    int kh   = lane >> 4;      // K-phase select

    int rowBase = blockIdx.x * 128 + wave * 16;
    int colBase = blockIdx.y * 32;
    int col0 = colBase + ln;

    if (tid < 32) { sSum[tid] = 0.f; sSq[tid] = 0.f; }
    // cooperative stage of W[:, colBase:colBase+32] into LDS (pair-interleaved)
    for (int idx = tid; idx < K * 32; idx += 256) {
        int k = idx >> 5;
        int c = idx & 31;
        sB[(k >> 1) * PSTRIDE + 2 * c + (k & 1)] = W[(size_t)k * Ncols + colBase + c];
    }
    __syncthreads();

    int arow = rowBase + ln;
    if (arow > M - 1) arow = M - 1;        // clamp; guarded on store/stats
    int aoff = arow * lda + 2 * kh;        // 32-bit offsets (<2^31)
    const float* sBp = sB + kh * PSTRIDE + 2 * ln;   // per-lane LDS base

    v8f acc0 = {0.f, 0.f, 0.f, 0.f, 0.f, 0.f, 0.f, 0.f};
    v8f acc1 = {0.f, 0.f, 0.f, 0.f, 0.f, 0.f, 0.f, 0.f};
    for (int k = 0; k < K; k += 4, aoff += 4) {
        int ka = k + 2 * kh;
        v2f a = *(const v2f*)(A + aoff);
        if (BN) {
            a.x = fmaxf(0.f, a.x * in_scale[ka]     + in_shift[ka]);
            a.y = fmaxf(0.f, a.y * in_scale[ka + 1] + in_shift[ka + 1]);
        }
        // pair index pa = k/2 + kh; b0 at cols [0,16), b1 at cols [16,32)
        v2f b0 = *(const v2f*)(sBp + (k >> 1) * PSTRIDE);
        v2f b1 = *(const v2f*)(sBp + (k >> 1) * PSTRIDE + 32);
        acc0 = __builtin_amdgcn_wmma_f32_16x16x4_f32(
            false, a, false, b0, (short)0, acc0, false, false);
        acc1 = __builtin_amdgcn_wmma_f32_16x16x4_f32(
            false, a, false, b1, (short)0, acc1, false, false);
    }

    float bc0 = bias ? bias[col0] : 0.f;
    float bc1 = bias ? bias[col0 + 16] : 0.f;
    float s0 = 0.f, q0 = 0.f, s1 = 0.f, q1 = 0.f;
#pragma unroll
    for (int v = 0; v < 8; ++v) {
        int row = rowBase + kh * 8 + v;    // C/D layout: M = v + 8*kh
        float v0 = acc0[v] + bc0;
        float v1 = acc1[v] + bc1;
        if (row < M) {
            int ob = row * Ncols + col0;
            Out[ob]      = v0;
            Out[ob + 16] = v1;
            s0 += v0; q0 += v0 * v0;
            s1 += v1; q1 += v1 * v1;
        }
    }
    if (colSum) {
        s0 += __shfl_xor(s0, 16, 32);      // partner lane holds same column
        q0 += __shfl_xor(q0, 16, 32);
        s1 += __shfl_xor(s1, 16, 32);
        q1 += __shfl_xor(q1, 16, 32);
        if (kh == 0) {
            atomicAdd(&sSum[ln], s0);      atomicAdd(&sSq[ln], q0);
            atomicAdd(&sSum[16 + ln], s1); atomicAdd(&sSq[16 + ln], q1);
        }
    }
    __syncthreads();
    if (colSum && tid < 32) {
        atomicAdd(&colSum[colBase + tid], sSum[tid]);
        atomicAdd(&colSq[colBase + tid], sSq[tid]);
    }
}

// ---------------------------------------------------------------------------
__global__ void k_stats(const float* __restrict__ sum, const float* __restrict__ sq,
                        const float* __restrict__ g, const float* __restrict__ b,
                        float* __restrict__ scale, float* __restrict__ shift,
                        float rows, int cols) {
    int c = blockIdx.x * blockDim.x + threadIdx.x;
    if (c >= cols) return;
    float m  = sum[c] / rows;
    float v  = sq[c] / rows - m * m;
    float sc = g[c] * rsqrtf(v + BN_EPS);
    scale[c] = sc;
    shift[c] = b[c] - m * sc;
}

// ---------------------------------------------------------------------------
__global__ void k_bn_act_pool(const float* __restrict__ U,
                              const float* __restrict__ scale,
                              const float* __restrict__ shift,
                              float* __restrict__ Hout, float* __restrict__ Pool,
                              const int* __restrict__ batch, int n, int relu) {
    long long t = (long long)blockIdx.x * blockDim.x + threadIdx.x;
    if (t >= (long long)n * 32) return;
    int i = (int)(t >> 5);
    int q = ((int)t & 31) * 4;
    float4 u  = *(const float4*)(U + (size_t)i * Dd + q);
    float4 sc = *(const float4*)(scale + q);
    float4 sh = *(const float4*)(shift + q);
    float4 v;
    v.x = u.x * sc.x + sh.x; v.y = u.y * sc.y + sh.y;
    v.z = u.z * sc.z + sh.z; v.w = u.w * sc.w + sh.w;
    if (relu) {
        v.x = fmaxf(v.x, 0.f); v.y = fmaxf(v.y, 0.f);
        v.z = fmaxf(v.z, 0.f); v.w = fmaxf(v.w, 0.f);
    }
    if (Hout) *(float4*)(Hout + (size_t)i * Dd + q) = v;
    if (Pool) {
        float* p = Pool + (size_t)batch[i] * Dd + q;
        atomicAdd(p + 0, v.x); atomicAdd(p + 1, v.y);
        atomicAdd(p + 2, v.z); atomicAdd(p + 3, v.w);
    }
}

// h[i] += vnf[batch[i]]
__global__ void k_add_vnf(float* __restrict__ h, const float* __restrict__ vnf,
                          const int* __restrict__ batch) {
    long long t = (long long)blockIdx.x * blockDim.x + threadIdx.x;
    if (t >= (long long)Nn * 32) return;
    int i = (int)(t >> 5);
    int q = ((int)t & 31) * 4;
    int g = batch[i];
    float4* hp = (float4*)(h + (size_t)i * Dd + q);
    float4 a = *hp;
    float4 b = *(const float4*)(vnf + (size_t)g * Dd + q);
    a.x += b.x; a.y += b.y; a.z += b.z; a.w += b.w;
    *hp = a;
}

__global__ void k_vinit(float* __restrict__ vnf, const float* __restrict__ vemb) {
    int t = blockIdx.x * blockDim.x + threadIdx.x;
    if (t < Gg * Dd) vnf[t] = vemb[t & (Dd - 1)];
}

__global__ void k_vadd(float* __restrict__ vin, const float* __restrict__ pooled,
                       const float* __restrict__ vnf) {
    int t = blockIdx.x * blockDim.x + threadIdx.x;
    if (t < Gg * Dd) vin[t] = pooled[t] + vnf[t];
}

// out[g,c] = readout[g,:] . Wc[:,c] + bc[c]
__global__ void k_final(const float* __restrict__ R, const float* __restrict__ Wc,
                        const float* __restrict__ bc, float* __restrict__ out) {
    int t = blockIdx.x * blockDim.x + threadIdx.x;
    if (t >= Gg * Cc) return;
    int g = t / Cc, c = t % Cc;
    float acc = bc[c];
    for (int k = 0; k < Dd; ++k) acc += R[(size_t)g * Dd + k] * Wc[k * Cc + c];
    out[t] = acc;
}

// ---------------------------------------------------------------------------
extern "C" void kernel_launch(void* const* d_in, const int* in_sizes, int n_in,
                              void* d_out, int out_size, void* d_ws, size_t ws_size,
                              hipStream_t stream) {
    (void)in_sizes; (void)n_in; (void)out_size; (void)ws_size;
    auto F = [&](int i) { return (const float*)d_in[i]; };

    const float* x     = F(0);
    const int*   src   = (const int*)d_in[1];
    const int*   dst   = src + Ee;
    const int*   batch = (const int*)d_in[2];
    // conv_mlps[l]: Wa, ba, ga, bga, Wb, bb  at 3 + 6*l
    // post_bns[l]: g, b at 21 + 2*l ; vemb=27 ; vmlp W1..bt2 = 28..35 ; Wc=36, bc=37
    const float* vemb = F(27);
    const float* Wc = F(36), *bc = F(37);

    float* ws = (float*)d_ws;
    size_t o = 0;
    float* h   = ws + o; o += (size_t)Nn * Dd;        // node features (post-act)
    float* agg = ws + o; o += (size_t)Nn * Dd;        // aggregation; aliased as u
    float* t   = ws + o; o += (size_t)Nn * 2 * Dd;    // hidden [N,256]
    float* u   = agg;                                  // GEMM2 raw out (agg is dead)
    float* sum1 = ws + o; o += 256;  float* sq1 = ws + o; o += 256;
    float* sc1  = ws + o; o += 256;  float* sh1 = ws + o; o += 256;
    float* sum2 = ws + o; o += 128;  float* sq2 = ws + o; o += 128;
    float* sc2  = ws + o; o += 128;  float* sh2 = ws + o; o += 128;
    float* vnf    = ws + o; o += Gg * Dd;
    float* pooled = ws + o; o += Gg * Dd;
    float* vin    = ws + o; o += Gg * Dd;
    float* rdout  = ws + o; o += Gg * Dd;

    const int TPB = 256;
    const int nb_nd  = (int)(((long long)Nn * 32 + TPB - 1) / TPB);
    const int nb_ed  = (int)(((long long)Ee * 32 + TPB - 1) / TPB);
    const int nb_cp  = (Nn * Dd / 4 + TPB - 1) / TPB;
    const int gx_big = (Nn + 127) / 128;
    const int gx_sm  = (Gg + 127) / 128;

    auto conv_layer = [&](int l, const float* hin, int relu, float* pool) {
        const float* Wa  = F(3 + 6 * l + 0), *ba  = F(3 + 6 * l + 1);
        const float* ga  = F(3 + 6 * l + 2), *bga = F(3 + 6 * l + 3);
        const float* Wb  = F(3 + 6 * l + 4), *bb  = F(3 + 6 * l + 5);
        const float* pg  = F(21 + 2 * l),    *pb  = F(21 + 2 * l + 1);

        hipLaunchKernelGGL(k_copy4, dim3(nb_cp), dim3(TPB), 0, stream, agg, hin, Nn * Dd / 4);
        hipLaunchKernelGGL(k_scatter, dim3(nb_ed), dim3(TPB), 0, stream, agg, hin, src, dst);
        hipMemsetAsync(sum1, 0, 512 * sizeof(float), stream);
        hipLaunchKernelGGL((k_gemm<false>), dim3(gx_big, 8), dim3(TPB), 0, stream,
                           agg, Dd, Wa, ba, (const float*)nullptr, (const float*)nullptr,
                           t, sum1, sq1, Nn, Dd, 2 * Dd);
        hipLaunchKernelGGL(k_stats, dim3(1), dim3(TPB), 0, stream,
                           sum1, sq1, ga, bga, sc1, sh1, (float)Nn, 2 * Dd);
        hipMemsetAsync(sum2, 0, 256 * sizeof(float), stream);
        hipLaunchKernelGGL((k_gemm<true>), dim3(gx_big, 4), dim3(TPB), 0, stream,
                           t, 2 * Dd, Wb, bb, sc1, sh1, u, sum2, sq2, Nn, 2 * Dd, Dd);
        hipLaunchKernelGGL(k_stats, dim3(1), dim3(TPB), 0, stream,
                           sum2, sq2, pg, pb, sc2, sh2, (float)Nn, Dd);
        hipLaunchKernelGGL(k_bn_act_pool, dim3(nb_nd), dim3(TPB), 0, stream,
                           u, sc2, sh2, (l == 2) ? (float*)nullptr : h, pool, batch, Nn, relu);
    };

    // ---- layer 0: conv -> BN -> ReLU
    conv_layer(0, x, /*relu=*/1, /*pool=*/nullptr);

    // ---- virtual node init; i = 0 iteration
    hipLaunchKernelGGL(k_vinit, dim3((Gg * Dd + TPB - 1) / TPB), dim3(TPB), 0, stream, vnf, vemb);
    hipLaunchKernelGGL(k_add_vnf, dim3(nb_nd), dim3(TPB), 0, stream, h, vnf, batch);
    hipMemsetAsync(pooled, 0, (size_t)Gg * Dd * sizeof(float), stream);
    conv_layer(1, h, /*relu=*/1, /*pool=*/pooled);

    // ---- virtual-node MLP: vnf = vmlp(pooled + vnf)   (reuses WMMA GEMM, M=G)
    hipLaunchKernelGGL(k_vadd, dim3((Gg * Dd + TPB - 1) / TPB), dim3(TPB), 0, stream,
                       vin, pooled, vnf);
    hipMemsetAsync(sum1, 0, 512 * sizeof(float), stream);
    hipLaunchKernelGGL((k_gemm<false>), dim3(gx_sm, 8), dim3(TPB), 0, stream,
                       vin, Dd, F(28), F(29), (const float*)nullptr, (const float*)nullptr,
                       t, sum1, sq1, Gg, Dd, 2 * Dd);
    hipLaunchKernelGGL(k_stats, dim3(1), dim3(TPB), 0, stream,
                       sum1, sq1, F(30), F(31), sc1, sh1, (float)Gg, 2 * Dd);
    hipMemsetAsync(sum2, 0, 256 * sizeof(float), stream);
    hipLaunchKernelGGL((k_gemm<true>), dim3(gx_sm, 4), dim3(TPB), 0, stream,
                       t, 2 * Dd, F(32), F(33), sc1, sh1, u, sum2, sq2, Gg, 2 * Dd, Dd);
    hipLaunchKernelGGL(k_stats, dim3(1), dim3(TPB), 0, stream,
                       sum2, sq2, F(34), F(35), sc2, sh2, (float)Gg, Dd);
    hipLaunchKernelGGL(k_bn_act_pool, dim3((Gg * 32 + TPB - 1) / TPB), dim3(TPB), 0, stream,
                       u, sc2, sh2, vnf, (float*)nullptr, batch, Gg, 1);

    // ---- i = 1 iteration: add vnf, conv(2), BN (no relu), readout pool
    hipLaunchKernelGGL(k_add_vnf, dim3(nb_nd), dim3(TPB), 0, stream, h, vnf, batch);
    hipMemsetAsync(rdout, 0, (size_t)Gg * Dd * sizeof(float), stream);
    conv_layer(2, h, /*relu=*/0, /*pool=*/rdout);

    // ---- classifier
    hipLaunchKernelGGL(k_final, dim3((Gg * Cc + TPB - 1) / TPB), dim3(TPB), 0, stream,
                       rdout, Wc, bc, (float*)d_out);
}